// RelativeAdditiveAttentionMultiHead_69020124447486
// MI455X (gfx1250) — compile-verified
//
#include <hip/hip_runtime.h>
#include <hip/hip_bf16.h>
#include <math.h>

typedef __attribute__((ext_vector_type(2))) float v2f;
typedef __attribute__((ext_vector_type(8))) float v8f;
typedef __attribute__((ext_vector_type(4))) unsigned int v4u;
typedef __attribute__((ext_vector_type(4))) int v4i;
typedef __attribute__((ext_vector_type(8))) int v8i;

#define TM1        127
#define KDIM       64
#define HC         4
#define CDIM       256
#define KV_STRIDE  68   // 64 + 4: row-to-row bank rotation -> conflict-free column reads
#define WK_STRIDE  68

#if __has_builtin(__builtin_amdgcn_tensor_load_to_lds) && __has_builtin(__builtin_amdgcn_s_wait_tensorcnt)
#define USE_TDM 1
#else
#define USE_TDM 0
#endif

// Hardware tanh (V_TANH_F32 is a CDNA5 TRANS op). Fallback: inline asm + v_nop
// to satisfy the 1-slot TRANS data hazard the compiler can't see through asm.
__device__ __forceinline__ float fast_tanh(float x) {
#if __has_builtin(__builtin_amdgcn_tanhf)
    return __builtin_amdgcn_tanhf(x);
#else
    float y;
    asm volatile("v_tanh_f32 %0, %1\n\tv_nop" : "=v"(y) : "v"(x));
    return y;
#endif
}

#if USE_TDM
// TDM: copy a rows x 64-float row-major tile Global->LDS, inserting 4 pad
// DWORDs after every 64 DWORDs (-> LDS row stride of 68 floats).
// D# bit layout per CDNA5 ISA ch.8 (groups 0/1; 2D tensor so groups 2/3 zero).
// This toolchain's builtin is the 6-arg form:
//   (uint32x4 g0, int32x8 g1, int32x4, int32x4, int32x8, i32 cpol)
__device__ __forceinline__ void tdm_load_rows64_padded(const float* gsrc,
                                                       unsigned lds_byte_addr,
                                                       int rows) {
    const unsigned long long ga = (unsigned long long)(size_t)gsrc;
    v4u g0;
    g0.x = 1u;                                              // count=1 (valid), user mode
    g0.y = lds_byte_addr;                                   // lds_addr [63:32]
    g0.z = (unsigned)(ga & 0xffffffffu);                    // global_addr[31:0]
    g0.w = (unsigned)((ga >> 32) & 0x01ffffffu)             // global_addr[56:32]
         | (2u << 30);                                      // type=2 ("image")
    v8i g1;
    g1[0] = (2 << 16)      // data_size = 4 bytes
          | (1 << 20)      // pad_enable
          | (5 << 22)      // pad_interval: every 64 DWORDs
          | (3 << 25);     // pad_amount: 4 DWORDs
    g1[1] = (64 << 16);             // tensor_dim0 = 64 elements  (bits 79:48)
    g1[2] = ((rows & 0xffff) << 16);// tensor_dim1[15:0]          (bits 111:80)
    g1[3] = (64 << 16);             // tile_dim0 = 64             (bits 127:112)
    g1[4] = rows;                   // tile_dim1                  (bits 143:128)
    g1[5] = 64;                     // tensor_dim0_stride = 64    (bits 207:160)
    g1[6] = 0;
    g1[7] = 0;
    const v4i zero4 = {0, 0, 0, 0};
    const v8i zero8 = {0, 0, 0, 0, 0, 0, 0, 0};
    __builtin_amdgcn_tensor_load_to_lds(g0, g1, zero4, zero4, zero8, 0);
}
#endif

__global__ __launch_bounds__(256)
void raa_mh_kernel(const float* __restrict__ q_x,
                   const float* __restrict__ kv_x,
                   const float* __restrict__ Wk,
                   const float* __restrict__ Wq,
                   const float* __restrict__ Wv,
                   const float* __restrict__ bias,
                   const float* __restrict__ Ws,
                   const float* __restrict__ bs,
                   float* __restrict__ out)
{
    __shared__ float kv_lds[128 * KV_STRIDE];    // kv slice, row 127 zero-padded
    __shared__ float wk_lds[CDIM * WK_STRIDE];   // Wk (row-major: [c][k])
    __shared__ float qb_lds[CDIM];               // q[c] + bias[c%64]
    __shared__ float ws_lds[KDIM];
    __shared__ float score_lds[HC * 128];
    __shared__ float w_lds[HC * 128];            // softmax weights
    __shared__ float z_lds[CDIM];                // z[h][k]

    (void)bs;  // additive scalar to every score -> cancels in softmax

    const int bt  = blockIdx.x;
    const int tid = threadIdx.x;
    const float* kv = kv_x + (size_t)bt * TM1 * KDIM;
    const float* qx = q_x + (size_t)bt * KDIM;

    // ---------------- Phase 0: stage kv, Wk (TDM), Ws; compute qb[c] ----------------
#if USE_TDM
    if (tid < 32) {  // one wave issues both DMAs (TDM is a per-wave instruction)
        tdm_load_rows64_padded(kv, (unsigned)(size_t)(void*)kv_lds, TM1);
        tdm_load_rows64_padded(Wk, (unsigned)(size_t)(void*)wk_lds, CDIM);
        __builtin_amdgcn_s_wait_tensorcnt(0);
    }
    if (tid < KDIM) kv_lds[TM1 * KV_STRIDE + tid] = 0.0f;   // zero pad row 127
#else
    for (int idx = tid; idx < 128 * KDIM; idx += 256) {
        const int row = idx >> 6, col = idx & 63;
        kv_lds[row * KV_STRIDE + col] = (row < TM1) ? kv[row * KDIM + col] : 0.0f;
    }
    for (int idx = tid; idx < CDIM * KDIM; idx += 256) {
        const int row = idx >> 6, col = idx & 63;
        wk_lds[row * WK_STRIDE + col] = Wk[idx];
    }
#endif
    if (tid < KDIM) ws_lds[tid] = Ws[tid];
    {
        const int c = tid;  // 0..255
        float s = bias[c & 63];
        const float4* wq4 = reinterpret_cast<const float4*>(Wq + c * KDIM);
        const float4* qx4 = reinterpret_cast<const float4*>(qx);
#pragma unroll
        for (int kk = 0; kk < 16; ++kk) {
            const float4 a = qx4[kk], w = wq4[kk];
            s += a.x * w.x + a.y * w.y + a.z * w.z + a.w * w.w;
        }
        qb_lds[c] = s;
    }
    __syncthreads();

    // ------- Phase 1: keys = kv @ Wk.T via f32 WMMA, fused hw-tanh + score -------
    const int wave = tid >> 5;
    const int lane = tid & 31;
    const int half = lane >> 4;   // 0: K+0/K+1 lanes, 1: K+2/K+3 lanes
    const int l15  = lane & 15;
    const int m0   = wave * 16;   // this wave owns key rows m0..m0+15

    // A fragments for the whole K=64 strip (16 k-steps of 4), kept in VGPRs.
    v2f afrag[16];
    {
        const int arow = m0 + l15;
#pragma unroll
        for (int kk = 0; kk < 16; ++kk) {
            const int k = kk * 4 + half * 2;
            v2f a;
            a.x = kv_lds[arow * KV_STRIDE + k];
            a.y = kv_lds[arow * KV_STRIDE + k + 1];
            afrag[kk] = a;
        }
    }

    for (int h = 0; h < HC; ++h) {
        float sacc[8] = {0.f, 0.f, 0.f, 0.f, 0.f, 0.f, 0.f, 0.f};
        for (int nsub = 0; nsub < 4; ++nsub) {
            const int c = h * 64 + nsub * 16 + l15;   // this lane's keys column
            v8f acc = {0.f, 0.f, 0.f, 0.f, 0.f, 0.f, 0.f, 0.f};
#pragma unroll
            for (int kk = 0; kk < 16; ++kk) {
                const int k = kk * 4 + half * 2;
                v2f b;                                 // B[k,c] = Wk[c,k]
                b.x = wk_lds[c * WK_STRIDE + k];
                b.y = wk_lds[c * WK_STRIDE + k + 1];
                acc = __builtin_amdgcn_wmma_f32_16x16x4_f32(
                        false, afrag[kk], false, b, (short)0, acc, false, false);
            }
            // acc[r] = keys[m0 + half*8 + r, c]; fuse tanh + Ws dot immediately.
            const float qb  = qb_lds[c];
            const float wsc = ws_lds[c & 63];
#pragma unroll
            for (int r = 0; r < 8; ++r)
                sacc[r] += wsc * fast_tanh(acc[r] + qb);
        }
        // Reduce over the 16 columns held by each lane-half.
#pragma unroll
        for (int r = 0; r < 8; ++r) {
            float v = sacc[r];
            v += __shfl_xor(v, 1, 32);
            v += __shfl_xor(v, 2, 32);
            v += __shfl_xor(v, 4, 32);
            v += __shfl_xor(v, 8, 32);
            sacc[r] = v;
        }
        if (l15 == 0) {
            const int mrow = m0 + half * 8;
#pragma unroll
            for (int r = 0; r < 8; ++r)
                score_lds[h * 128 + mrow + r] = sacc[r];
        }
    }
    __syncthreads();

    // ---------------- Phase 2: softmax over the 127 keys, per head ----------------
    if (wave < HC) {
        const int h = wave;
        float vals[4];
        float mx = -INFINITY;
#pragma unroll
        for (int i = 0; i < 4; ++i) {
            const int n = lane + i * 32;
            vals[i] = (n < TM1) ? score_lds[h * 128 + n] : -INFINITY;
            mx = fmaxf(mx, vals[i]);
        }
        mx = fmaxf(mx, __shfl_xor(mx, 1, 32));
        mx = fmaxf(mx, __shfl_xor(mx, 2, 32));
        mx = fmaxf(mx, __shfl_xor(mx, 4, 32));
        mx = fmaxf(mx, __shfl_xor(mx, 8, 32));
        mx = fmaxf(mx, __shfl_xor(mx, 16, 32));
        float sum = 0.f;
#pragma unroll
        for (int i = 0; i < 4; ++i) {
            const float e = __expf(vals[i] - mx);   // exp(-inf) = 0 for pad row
            vals[i] = e;
            sum += e;
        }
        sum += __shfl_xor(sum, 1, 32);
        sum += __shfl_xor(sum, 2, 32);
        sum += __shfl_xor(sum, 4, 32);
        sum += __shfl_xor(sum, 8, 32);
        sum += __shfl_xor(sum, 16, 32);
        const float inv = 1.0f / sum;
#pragma unroll
        for (int i = 0; i < 4; ++i) {
            const int n = lane + i * 32;
            w_lds[h * 128 + n] = vals[i] * inv;
        }
    }
    __syncthreads();

    // -------- Phase 3: z[h,k] = sum_n w[n,h] * kv[n,k]  (values-GEMM eliminated) --------
    {
        const int h = tid >> 6, k = tid & 63;
        float z = 0.f;
        for (int n = 0; n < TM1; ++n)
            z += w_lds[h * 128 + n] * kv_lds[n * KV_STRIDE + k];
        z_lds[tid] = z;   // tid == h*64 + k
    }
    __syncthreads();

    // -------- Phase 4: out[c] = sum_k z[h,k] * Wv[c,k] --------
    {
        const int c = tid;
        const int h = c >> 6;
        const float4* wv4 = reinterpret_cast<const float4*>(Wv + c * KDIM);
        float acc = 0.f;
#pragma unroll
        for (int kk = 0; kk < 16; ++kk) {
            const float4 w = wv4[kk];
            acc += z_lds[h * 64 + kk * 4 + 0] * w.x
                 + z_lds[h * 64 + kk * 4 + 1] * w.y
                 + z_lds[h * 64 + kk * 4 + 2] * w.z
                 + z_lds[h * 64 + kk * 4 + 3] * w.w;
        }
        out[(size_t)bt * CDIM + c] = acc;
    }
}

extern "C" void kernel_launch(void* const* d_in, const int* in_sizes, int n_in,
                              void* d_out, int out_size, void* d_ws, size_t ws_size,
                              hipStream_t stream) {
    const float* q_x  = (const float*)d_in[0];
    const float* kv_x = (const float*)d_in[1];
    const float* Wk   = (const float*)d_in[2];
    const float* Wq   = (const float*)d_in[3];
    const float* Wv   = (const float*)d_in[4];
    const float* bias = (const float*)d_in[5];
    const float* Ws   = (const float*)d_in[6];
    const float* bs   = (const float*)d_in[7];
    float* out = (float*)d_out;

    const int bt = in_sizes[0] / KDIM;   // b*t = 1024 blocks, one per (b,t)
    raa_mh_kernel<<<bt, 256, 0, stream>>>(q_x, kv_x, Wk, Wq, Wv, bias, Ws, bs, out);
}